// GCN3D_Apr14_PoolingNoFc_51281909514720
// MI455X (gfx1250) — compile-verified
//
#include <hip/hip_runtime.h>
#include <hip/hip_bf16.h>

typedef __attribute__((ext_vector_type(2))) float v2f;
typedef __attribute__((ext_vector_type(8))) float v8f;

#define NN      20000
#define NPG     10000
#define NB      2
#define NE      640000
#define CLUST   128
#define NP      256      // pooled nodes
#define F1      64
#define F2      256
#define EPSV    1e-5f

static __device__ __forceinline__ float elu_f(float v) {
    return v > 0.f ? v : (__expf(v) - 1.f);
}

// ---------------- generic fp32 WMMA GEMM: C[M,N] = A[M,K] @ B[K,N] (+bias, +elu) ----
// one wave computes one 16x16 tile, K stepped by 4 via V_WMMA_F32_16X16X4_F32
__global__ void wmma_gemm_f32(const float* __restrict__ A, const float* __restrict__ B,
                              float* __restrict__ C, int M, int N, int K,
                              const float* __restrict__ bias, int do_elu) {
    int wave = threadIdx.x >> 5;
    int lane = threadIdx.x & 31;
    int tiles_n = N >> 4;
    int tiles = (M >> 4) * tiles_n;
    int tile = blockIdx.x * (blockDim.x >> 5) + wave;
    if (tile >= tiles) return;                 // uniform per wave: EXEC stays all-1s
    int tm = tile / tiles_n, tn = tile - tm * tiles_n;
    int hi = lane >> 4, lo = lane & 15;
    int arow = tm * 16 + lo;                   // A fragment row (M)
    int col  = tn * 16 + lo;                   // B/C/D column (N)

    v8f acc = {};
    const float* Arow = A + (long long)arow * K;
    for (int k0 = 0; k0 < K; k0 += 4) {
        int ka = k0 + 2 * hi;
        v2f a, b;
        a.x = Arow[ka];
        a.y = Arow[ka + 1];
        b.x = B[(long long)ka * N + col];
        b.y = B[(long long)(ka + 1) * N + col];
        acc = __builtin_amdgcn_wmma_f32_16x16x4_f32(false, a, false, b,
                                                    (short)0, acc, false, false);
    }
#pragma unroll
    for (int i = 0; i < 8; ++i) {
        int r = tm * 16 + i + 8 * hi;
        float v = acc[i];
        if (bias) v += bias[col];
        if (do_elu) v = elu_f(v);
        C[(long long)r * N + col] = v;
    }
}

// ---------------- small scalar pieces -------------------------------------------
__global__ void fill_kernel(float* p, float v, int n) {
    int i = blockIdx.x * blockDim.x + threadIdx.x;
    if (i < n) p[i] = v;
}

__global__ void deg_edge_kernel(const int* __restrict__ cols, float* __restrict__ deg, int ne) {
    int e = blockIdx.x * blockDim.x + threadIdx.x;
    if (e < ne) atomicAdd(&deg[cols[e]], 1.0f);
}

__global__ void rsqrt_kernel(float* d, int n) {
    int i = blockIdx.x * blockDim.x + threadIdx.x;
    if (i < n) { float v = d[i]; d[i] = v > 0.f ? rsqrtf(v) : 0.f; }
}

// xw1 = x[NN,3] @ W_g1[3,64]
__global__ void xw1_kernel(const float* __restrict__ x, const float* __restrict__ W,
                           float* __restrict__ out, int nn) {
    int idx = blockIdx.x * blockDim.x + threadIdx.x;
    if (idx >= nn * F1) return;
    int n = idx >> 6, f = idx & 63;
    const float* xr = x + n * 3;
    out[idx] = xr[0] * W[f] + xr[1] * W[F1 + f] + xr[2] * W[2 * F1 + f];
}

// edge scatter: acc[col] += dis[row]*dis[col] * xw[row]; 4 features per thread
__global__ void scatter_kernel(const float* __restrict__ xw, const int* __restrict__ rows,
                               const int* __restrict__ cols, const float* __restrict__ dis,
                               float* __restrict__ acc, int ne, int F) {
    int chunks = F >> 2;
    long long idx = (long long)blockIdx.x * blockDim.x + threadIdx.x;
    if (idx >= (long long)ne * chunks) return;
    int e = (int)(idx / chunks);
    int c = (int)(idx - (long long)e * chunks) * 4;
    int r = rows[e], cl = cols[e];
    float nm = dis[r] * dis[cl];
    const float4 v = *(const float4*)(xw + (long long)r * F + c);
    float* o = acc + (long long)cl * F + c;
    atomicAdd(o + 0, nm * v.x);
    atomicAdd(o + 1, nm * v.y);
    atomicAdd(o + 2, nm * v.z);
    atomicAdd(o + 3, nm * v.w);
}

// h = elu(acc + dis^2 * xw + bias)  (self-loop term + bias + ELU), in place on acc
__global__ void finalize_kernel(const float* __restrict__ xw, const float* __restrict__ dis,
                                const float* __restrict__ bias, float* __restrict__ acc,
                                int nn, int F) {
    long long idx = (long long)blockIdx.x * blockDim.x + threadIdx.x;
    if (idx >= (long long)nn * F) return;
    int n = (int)(idx / F), f = (int)(idx - (long long)n * F);
    float d = dis[n];
    float v = acc[idx] + d * d * xw[idx] + bias[f];
    acc[idx] = elu_f(v);
}

// instance-norm stats: one block per (graph b, feature f)
__global__ void in_stats_kernel(const float* __restrict__ h, const int* __restrict__ in_batch,
                                float* __restrict__ mean, float* __restrict__ rstd,
                                int nn, int F) {
    int b = blockIdx.x / F, f = blockIdx.x - b * F;
    float s = 0.f, ss = 0.f, cnt = 0.f;
    for (int n = threadIdx.x; n < nn; n += blockDim.x) {
        if (in_batch[n] == b) {
            float v = h[(long long)n * F + f];
            s += v; ss += v * v; cnt += 1.f;
        }
    }
    __shared__ float r0[256], r1[256], r2[256];
    int tid = threadIdx.x;
    r0[tid] = s; r1[tid] = ss; r2[tid] = cnt;
    __syncthreads();
    for (int st = 128; st > 0; st >>= 1) {
        if (tid < st) { r0[tid] += r0[tid + st]; r1[tid] += r1[tid + st]; r2[tid] += r2[tid + st]; }
        __syncthreads();
    }
    if (tid == 0) {
        float c = r2[0] > 0.f ? r2[0] : 1.f;
        float m = r0[0] / c;
        mean[b * F + f] = m;
        rstd[b * F + f] = rsqrtf(r1[0] / c - m * m + EPSV);
    }
}

__global__ void bc_kernel(const int* __restrict__ cluster, const int* __restrict__ in_batch,
                          int* __restrict__ bc, int nn) {
    int n = blockIdx.x * blockDim.x + threadIdx.x;
    if (n < nn) bc[n] = cluster[n] + in_batch[n] * CLUST;
}

// normalize + pool: px_acc[bc[n]] += (h - mean)*rstd ; cnt_c[bc[n]] += 1
__global__ void norm_pool_kernel(const float* __restrict__ h, const int* __restrict__ in_batch,
                                 const int* __restrict__ bc, const float* __restrict__ mean,
                                 const float* __restrict__ rstd, float* __restrict__ px_acc,
                                 float* __restrict__ cnt_c, int nn, int F) {
    long long idx = (long long)blockIdx.x * blockDim.x + threadIdx.x;
    if (idx >= (long long)nn * F) return;
    int n = (int)(idx / F), f = (int)(idx - (long long)n * F);
    int b = in_batch[n];
    float v = (h[idx] - mean[b * F + f]) * rstd[b * F + f];
    int p = bc[n];
    atomicAdd(&px_acc[(long long)p * F + f], v);
    if (f == 0) atomicAdd(&cnt_c[p], 1.f);
}

__global__ void px_div_kernel(float* __restrict__ px, const float* __restrict__ cnt_c, int total, int F) {
    int i = blockIdx.x * blockDim.x + threadIdx.x;
    if (i < total) px[i] /= fmaxf(cnt_c[i / F], 1.f);
}

__global__ void adj_build_kernel(const int* __restrict__ rows, const int* __restrict__ cols,
                                 const int* __restrict__ bc, float* __restrict__ A, int ne) {
    int e = blockIdx.x * blockDim.x + threadIdx.x;
    if (e < ne) A[bc[cols[e]] * NP + bc[rows[e]]] = 1.0f;   // races all write 1.0
}

// per-row rowsum of A_hat (diag forced to 1) -> disc[i] = rsqrt(rowsum)
__global__ void adj_deg_kernel(const float* __restrict__ A, float* __restrict__ disc) {
    int i = blockIdx.x;          // row
    int j = threadIdx.x;         // col
    float v = (i == j) ? 1.0f : A[i * NP + j];
    __shared__ float r[NP];
    r[j] = v;
    __syncthreads();
    for (int st = NP / 2; st > 0; st >>= 1) {
        if (j < st) r[j] += r[j + st];
        __syncthreads();
    }
    if (j == 0) disc[i] = rsqrtf(r[0]);
}

__global__ void adj_norm_kernel(const float* __restrict__ A, const float* __restrict__ disc,
                                float* __restrict__ An) {
    int idx = blockIdx.x * blockDim.x + threadIdx.x;
    if (idx >= NP * NP) return;
    int i = idx / NP, j = idx - i * NP;
    float a = (i == j) ? 1.0f : A[idx];
    An[idx] = disc[i] * a * disc[j];
}

// small scalar GEMM (for N=5 tails): C[M,N] = A[M,K]@B[K,N] (+bias)
__global__ void gemm_small_kernel(const float* __restrict__ A, const float* __restrict__ B,
                                  float* __restrict__ C, int M, int N, int K,
                                  const float* __restrict__ bias) {
    int idx = blockIdx.x * blockDim.x + threadIdx.x;
    if (idx >= M * N) return;
    int m = idx / N, n = idx - m * N;
    float s = bias ? bias[n] : 0.f;
    for (int k = 0; k < K; ++k) s += A[m * K + k] * B[k * N + n];
    C[idx] = s;
}

// o_feat = W_fc[n_out, K] @ g[K]; one wave per output row, float4 streaming
__global__ void fc_gemv_kernel(const float* __restrict__ W, const float* __restrict__ g,
                               float* __restrict__ out, int n_out, int K) {
    __shared__ __align__(16) float gs[1280];
    for (int i = threadIdx.x; i < K; i += blockDim.x) gs[i] = g[i];
    __syncthreads();
    int wave = threadIdx.x >> 5, lane = threadIdx.x & 31;
    int row = blockIdx.x * (blockDim.x >> 5) + wave;
    if (row >= n_out) return;
    const float4* Wr = (const float4*)(W + (long long)row * K);
    const float4* G4 = (const float4*)gs;
    float acc = 0.f;
    int nchunk = K >> 7;   // K / (4*32)
    for (int t = 0; t < nchunk; ++t) {
        float4 w = Wr[t * 32 + lane];
        float4 gg = G4[t * 32 + lane];
        acc += w.x * gg.x + w.y * gg.y + w.z * gg.z + w.w * gg.w;
    }
    for (int off = 16; off > 0; off >>= 1) acc += __shfl_xor(acc, off, 32);
    if (lane == 0) out[row] = acc;
}

static inline unsigned cdiv(long long a, long long b) { return (unsigned)((a + b - 1) / b); }

extern "C" void kernel_launch(void* const* d_in, const int* in_sizes, int n_in,
                              void* d_out, int out_size, void* d_ws, size_t ws_size,
                              hipStream_t stream) {
    const float* x        = (const float*)d_in[0];
    const int*   eidx     = (const int*)d_in[1];
    const int*   in_batch = (const int*)d_in[2];
    const int*   cluster  = (const int*)d_in[3];
    const float* W_g1 = (const float*)d_in[4];
    const float* b_g1 = (const float*)d_in[5];
    const float* W_g2 = (const float*)d_in[6];
    const float* b_g2 = (const float*)d_in[7];
    const float* W_l1 = (const float*)d_in[8];
    const float* b_l1 = (const float*)d_in[9];
    const float* W_l2 = (const float*)d_in[10];
    const float* b_l2 = (const float*)d_in[11];
    const float* W_m1 = (const float*)d_in[12];
    const float* b_m1 = (const float*)d_in[13];
    const float* W_o1 = (const float*)d_in[14];
    const float* b_o1 = (const float*)d_in[15];
    const float* W_fc = (const float*)d_in[16];

    const int ne = in_sizes[1] / 2;
    const int* erows = eidx;
    const int* ecols = eidx + ne;

    float* out = (float*)d_out;
    float* g_feat = out;                 // [NP*5] = 1280
    float* o_feat = out + NP * 5;        // [60000]
    const int fc_k = NP * 5;             // 1280
    const int n_out = out_size - NP * 5; // 60000

    // ---- workspace layout (fp32 slots) ----
    float* ws   = (float*)d_ws;
    float* dis  = ws;                       // 20000 (deg -> rsqrt in place)
    float* xw1  = dis + NN;                 // NN*F1
    float* acc1 = xw1 + (long long)NN * F1; // NN*F1  (-> h1)
    float* xw2  = acc1 + (long long)NN * F1;// NN*F2
    float* acc2 = xw2 + (long long)NN * F2; // NN*F2  (-> h2)
    float* mean = acc2 + (long long)NN * F2;// NB*F2
    float* rstd = mean + NB * F2;           // NB*F2
    int*   bc   = (int*)(rstd + NB * F2);   // NN ints
    float* cntc = (float*)(bc + NN);        // NP
    float* px   = cntc + NP;                // NP*F2
    float* A    = px + NP * F2;             // NP*NP
    float* disc = A + NP * NP;              // NP
    float* An   = disc + NP;                // NP*NP
    float* t1   = An + NP * NP;             // NP*128
    float* z1   = t1 + NP * 128;            // NP*128
    float* t2   = z1 + NP * 128;            // NP*64
    float* z2   = t2 + NP * 64;             // NP*64
    float* t3   = z2 + NP * 64;             // NP*16
    float* z3   = t3 + NP * 16;             // NP*16
    float* g4   = z3 + NP * 16;             // NP*5

    const int TB = 256;

    // ---- zero accumulators (poisoned workspace) ----
    hipMemsetAsync(acc1, 0, (size_t)NN * F1 * sizeof(float), stream);
    hipMemsetAsync(acc2, 0, (size_t)NN * F2 * sizeof(float), stream);
    hipMemsetAsync(px,   0, (size_t)NP * F2 * sizeof(float), stream);
    hipMemsetAsync(cntc, 0, (size_t)NP * sizeof(float), stream);
    hipMemsetAsync(A,    0, (size_t)NP * NP * sizeof(float), stream);

    // ---- degrees (self-loop => start at 1) ----
    fill_kernel<<<cdiv(NN, TB), TB, 0, stream>>>(dis, 1.0f, NN);
    deg_edge_kernel<<<cdiv(ne, TB), TB, 0, stream>>>(ecols, dis, ne);
    rsqrt_kernel<<<cdiv(NN, TB), TB, 0, stream>>>(dis, NN);

    // ---- GCN layer 1 (3 -> 64) ----
    xw1_kernel<<<cdiv((long long)NN * F1, TB), TB, 0, stream>>>(x, W_g1, xw1, NN);
    scatter_kernel<<<cdiv((long long)ne * (F1 / 4), TB), TB, 0, stream>>>(xw1, erows, ecols, dis, acc1, ne, F1);
    finalize_kernel<<<cdiv((long long)NN * F1, TB), TB, 0, stream>>>(xw1, dis, b_g1, acc1, NN, F1);

    // ---- GCN layer 2 (64 -> 256): WMMA GEMM then scatter ----
    {
        int tiles = (NN / 16) * (F2 / 16);  // 20000 tiles
        wmma_gemm_f32<<<cdiv(tiles, 8), TB, 0, stream>>>(acc1, W_g2, xw2, NN, F2, F1, nullptr, 0);
    }
    scatter_kernel<<<cdiv((long long)ne * (F2 / 4), TB), TB, 0, stream>>>(xw2, erows, ecols, dis, acc2, ne, F2);
    finalize_kernel<<<cdiv((long long)NN * F2, TB), TB, 0, stream>>>(xw2, dis, b_g2, acc2, NN, F2);

    // ---- instance norm + avg pool ----
    in_stats_kernel<<<NB * F2, TB, 0, stream>>>(acc2, in_batch, mean, rstd, NN, F2);
    bc_kernel<<<cdiv(NN, TB), TB, 0, stream>>>(cluster, in_batch, bc, NN);
    norm_pool_kernel<<<cdiv((long long)NN * F2, TB), TB, 0, stream>>>(acc2, in_batch, bc, mean, rstd, px, cntc, NN, F2);
    px_div_kernel<<<cdiv(NP * F2, TB), TB, 0, stream>>>(px, cntc, NP * F2, F2);

    // ---- coarse adjacency ----
    adj_build_kernel<<<cdiv(ne, TB), TB, 0, stream>>>(erows, ecols, bc, A, ne);
    adj_deg_kernel<<<NP, NP, 0, stream>>>(A, disc);
    adj_norm_kernel<<<cdiv(NP * NP, TB), TB, 0, stream>>>(A, disc, An);

    // ---- dense coarse GCN stack (all WMMA) ----
    // z1 = elu(An @ (px @ W_l1) + b_l1)   256x256x128
    wmma_gemm_f32<<<cdiv((NP/16)*(128/16), 8), TB, 0, stream>>>(px, W_l1, t1, NP, 128, 256, nullptr, 0);
    wmma_gemm_f32<<<cdiv((NP/16)*(128/16), 8), TB, 0, stream>>>(An, t1, z1, NP, 128, NP, b_l1, 1);
    // z2 = elu(An @ (z1 @ W_l2) + b_l2)   -> 64
    wmma_gemm_f32<<<cdiv((NP/16)*(64/16), 8), TB, 0, stream>>>(z1, W_l2, t2, NP, 64, 128, nullptr, 0);
    wmma_gemm_f32<<<cdiv((NP/16)*(64/16), 8), TB, 0, stream>>>(An, t2, z2, NP, 64, NP, b_l2, 1);
    // z3 = elu(An @ (z2 @ W_m1) + b_m1)   -> 16
    wmma_gemm_f32<<<cdiv((NP/16)*(16/16), 8), TB, 0, stream>>>(z2, W_m1, t3, NP, 16, 64, nullptr, 0);
    wmma_gemm_f32<<<cdiv((NP/16)*(16/16), 8), TB, 0, stream>>>(An, t3, z3, NP, 16, NP, b_m1, 1);
    // g_feat = (An @ (z3 @ W_o1) + b_o1).reshape(-1)   (N=5 -> scalar)
    gemm_small_kernel<<<cdiv(NP * 5, TB), TB, 0, stream>>>(z3, W_o1, g4, NP, 5, 16, nullptr);
    gemm_small_kernel<<<cdiv(NP * 5, TB), TB, 0, stream>>>(An, g4, g_feat, NP, 5, NP, b_o1);

    // ---- o_feat = W_fc @ g_feat : HBM-bound GEMV, wave per row ----
    fc_gemv_kernel<<<cdiv(n_out, TB / 32), TB, 0, stream>>>(W_fc, g_feat, o_feat, n_out, fc_k);
}